// TransitionDown_44332652430156
// MI455X (gfx1250) — compile-verified
//
#include <hip/hip_runtime.h>

#define BB 8
#define NN 8192
#define CIN 64
#define COUT 128
#define MM 2048
#define KNN 16
#define BN_EPS 1e-5f

typedef __attribute__((ext_vector_type(2))) float v2f;
typedef __attribute__((ext_vector_type(8))) float v8f;

// ---------------------------------------------------------------------------
// 1) Farthest point sampling: one 1024-thread block per batch.
//    Each thread owns 8 points in registers; per-iteration block argmax via LDS.
//    Writes fps indices (ws) and p2 (output region) directly.
// ---------------------------------------------------------------------------
__global__ __launch_bounds__(1024) void fps_kernel(const float* __restrict__ p1,
                                                   int* __restrict__ fpsIdx,
                                                   float* __restrict__ p2out) {
  const int b = blockIdx.x;
  const int t = threadIdx.x;
  const float* pb = p1 + (size_t)b * NN * 3;

  __shared__ float svals[1024];
  __shared__ int   sidx[1024];
  __shared__ float selx, sely, selz;

  float px[8], py[8], pz[8], md[8];
#pragma unroll
  for (int s = 0; s < 8; ++s) {
    int j = t + s * 1024;
    px[s] = pb[j * 3 + 0];
    py[s] = pb[j * 3 + 1];
    pz[s] = pb[j * 3 + 2];
    md[s] = 1e10f;
  }

  if (t == 0) {
    fpsIdx[b * MM + 0] = 0;
    selx = pb[0]; sely = pb[1]; selz = pb[2];
    float* o = p2out + (size_t)(b * MM + 0) * 3;
    o[0] = pb[0]; o[1] = pb[1]; o[2] = pb[2];
  }

  for (int it = 1; it < MM; ++it) {
    __syncthreads();               // makes sel{x,y,z} from prev iter visible
    const float sx = selx, sy = sely, sz = selz;
    float best = -1.0f;
    int bestj = 0;
#pragma unroll
    for (int s = 0; s < 8; ++s) {
      float dx = px[s] - sx, dy = py[s] - sy, dz = pz[s] - sz;
      float d = dx * dx + dy * dy + dz * dz;
      float m = fminf(md[s], d);
      md[s] = m;
      if (m > best) { best = m; bestj = t + s * 1024; }
    }
    svals[t] = best;
    sidx[t] = bestj;
    __syncthreads();
    for (int off = 512; off > 0; off >>= 1) {
      if (t < off) {
        float v = svals[t + off]; int i = sidx[t + off];
        float v0 = svals[t];      int i0 = sidx[t];
        if (v > v0 || (v == v0 && i < i0)) { svals[t] = v; sidx[t] = i; }
      }
      __syncthreads();
    }
    if (t == 0) {
      int nx = sidx[0];
      fpsIdx[b * MM + it] = nx;
      float ax = pb[nx * 3 + 0], ay = pb[nx * 3 + 1], az = pb[nx * 3 + 2];
      selx = ax; sely = ay; selz = az;
      float* o = p2out + (size_t)(b * MM + it) * 3;
      o[0] = ax; o[1] = ay; o[2] = az;
    }
  }
}

// ---------------------------------------------------------------------------
// 2) h = x @ W^T via V_WMMA_F32_16X16X4_F32.
//    Block = 256 threads = 8 waves; block computes a 16-row stripe of h,
//    wave w computes columns [16w, 16w+16). LDS rows padded to 65 floats to
//    dodge 64-bank conflicts on stride-64 fragment reads.
// ---------------------------------------------------------------------------
__global__ __launch_bounds__(256) void gemm_wmma_kernel(const float* __restrict__ x,
                                                        const float* __restrict__ W,
                                                        float* __restrict__ h) {
  __shared__ float ldsx[16 * 65];
  __shared__ float ldsw[COUT * 65];

  const int t = threadIdx.x;
  const int lane = t & 31;
  const int wave = t >> 5;
  const size_t rowBase = (size_t)blockIdx.x * 16;
  const float* xg = x + rowBase * CIN;   // 16x64 tile is contiguous

  for (int i = t; i < 16 * CIN; i += 256)
    ldsx[(i >> 6) * 65 + (i & 63)] = xg[i];
  for (int i = t; i < COUT * CIN; i += 256)
    ldsw[(i >> 6) * 65 + (i & 63)] = W[i];
  __syncthreads();

  const int half = lane >> 4;        // 0: lanes 0-15 (K=0,1), 1: lanes 16-31 (K=2,3)
  const int l16  = lane & 15;
  const int arow = l16;              // A: M = lane%16
  const int bcol = wave * 16 + l16;  // B: N column this lane feeds

  v8f acc = {};
#pragma unroll
  for (int kk = 0; kk < CIN / 4; ++kk) {
    const int kb = kk * 4 + half * 2;
    v2f a, bf;
    a.x  = ldsx[arow * 65 + kb];
    a.y  = ldsx[arow * 65 + kb + 1];
    bf.x = ldsw[bcol * 65 + kb];     // B[k][n] = W[n][k]
    bf.y = ldsw[bcol * 65 + kb + 1];
    acc = __builtin_amdgcn_wmma_f32_16x16x4_f32(false, a, false, bf,
                                                (short)0, acc, false, false);
  }

  const int rowOff = half * 8;       // C/D: VGPR r -> row r (lo half) / r+8 (hi half)
#pragma unroll
  for (int r = 0; r < 8; ++r)
    h[(rowBase + r + rowOff) * COUT + bcol] = acc[r];
}

// ---------------------------------------------------------------------------
// 3) BatchNorm batch statistics: one block per channel; h is L2-resident.
// ---------------------------------------------------------------------------
__global__ __launch_bounds__(256) void bnstats_kernel(const float* __restrict__ h,
                                                      float* __restrict__ meanArr,
                                                      float* __restrict__ rstdArr) {
  const int c = blockIdx.x;
  const int t = threadIdx.x;
  __shared__ float ss[256], ss2[256];
  float s = 0.f, s2 = 0.f;
  for (int i = t; i < BB * NN; i += 256) {
    float v = h[(size_t)i * COUT + c];
    s += v; s2 += v * v;
  }
  ss[t] = s; ss2[t] = s2;
  __syncthreads();
  for (int off = 128; off > 0; off >>= 1) {
    if (t < off) { ss[t] += ss[t + off]; ss2[t] += ss2[t + off]; }
    __syncthreads();
  }
  if (t == 0) {
    const float inv = 1.0f / (float)(BB * NN);
    float mean = ss[0] * inv;
    float var  = ss2[0] * inv - mean * mean;   // ddof=0, matches jnp.var
    meanArr[c] = mean;
    rstdArr[c] = rsqrtf(var + BN_EPS);
  }
}

// ---------------------------------------------------------------------------
// 4) kNN: one thread per query, register-resident sorted top-16,
//    p1 staged through LDS in 1024-point (12 KB) chunks.
// ---------------------------------------------------------------------------
__global__ __launch_bounds__(128) void knn_kernel(const float* __restrict__ p1,
                                                  const float* __restrict__ p2,
                                                  int* __restrict__ nbr) {
  const int qBlock = blockIdx.x;                 // B * (M/128) blocks
  const int b = qBlock / (MM / 128);
  const int m = (qBlock % (MM / 128)) * 128 + threadIdx.x;
  const int t = threadIdx.x;
  const float* pb = p1 + (size_t)b * NN * 3;
  const float* q  = p2 + (size_t)(b * MM + m) * 3;
  const float qx = q[0], qy = q[1], qz = q[2];

  __shared__ float lp[1024 * 3];
  float dk[KNN]; int ik[KNN];
#pragma unroll
  for (int k = 0; k < KNN; ++k) { dk[k] = 3.4e38f; ik[k] = 0; }

  for (int chunk = 0; chunk < NN; chunk += 1024) {
    __syncthreads();
    for (int i = t; i < 1024 * 3; i += 128) lp[i] = pb[(size_t)chunk * 3 + i];
    __syncthreads();
    for (int j = 0; j < 1024; ++j) {           // all lanes same addr: LDS broadcast
      float dx = qx - lp[j * 3 + 0];
      float dy = qy - lp[j * 3 + 1];
      float dz = qz - lp[j * 3 + 2];
      float d = dx * dx + dy * dy + dz * dz;
      if (d < dk[KNN - 1]) {                   // strict < keeps earlier index on ties
        dk[KNN - 1] = d; ik[KNN - 1] = chunk + j;
#pragma unroll
        for (int k = KNN - 1; k >= 1; --k) {
          if (dk[k] < dk[k - 1]) {
            float td = dk[k]; dk[k] = dk[k - 1]; dk[k - 1] = td;
            int   ti = ik[k]; ik[k] = ik[k - 1]; ik[k - 1] = ti;
          }
        }
      }
    }
  }
  int* o = nbr + (size_t)(b * MM + m) * KNN;
#pragma unroll
  for (int k = 0; k < KNN; ++k) o[k] = ik[k];
}

// ---------------------------------------------------------------------------
// 5) Gather neighbor features, apply BN affine + ReLU, max over K.
//    One block per (b,m); thread = channel; gathers are 512B-coalesced rows.
// ---------------------------------------------------------------------------
__global__ __launch_bounds__(128) void gather_bn_max_kernel(
    const float* __restrict__ h, const int* __restrict__ nbr,
    const float* __restrict__ meanArr, const float* __restrict__ rstdArr,
    const float* __restrict__ gamma, const float* __restrict__ beta,
    float* __restrict__ y) {
  const int bm = blockIdx.x;                  // 0 .. B*M-1
  const int b  = bm / MM;
  const int c  = threadIdx.x;
  __shared__ int snb[KNN];
  if (c < KNN) snb[c] = nbr[(size_t)bm * KNN + c];
  __syncthreads();
  const float mean = meanArr[c], rstd = rstdArr[c];
  const float g = gamma[c], be = beta[c];
  float best = -3.4e38f;
#pragma unroll
  for (int k = 0; k < KNN; ++k) {
    const int j = snb[k];
    float v = h[((size_t)b * NN + j) * COUT + c];
    v = g * (v - mean) * rstd + be;           // BN before max (gamma may be <0)
    v = fmaxf(v, 0.0f);
    best = fmaxf(best, v);
  }
  y[(size_t)bm * COUT + c] = best;
}

// ---------------------------------------------------------------------------
extern "C" void kernel_launch(void* const* d_in, const int* in_sizes, int n_in,
                              void* d_out, int out_size, void* d_ws, size_t ws_size,
                              hipStream_t stream) {
  (void)in_sizes; (void)n_in; (void)out_size; (void)ws_size;
  const float* x     = (const float*)d_in[0];  // (B,N,Cin)
  const float* p1    = (const float*)d_in[1];  // (B,N,3)
  const float* W     = (const float*)d_in[2];  // (Cout,Cin)
  const float* gamma = (const float*)d_in[3];  // (Cout,)
  const float* beta  = (const float*)d_in[4];  // (Cout,)

  float* y  = (float*)d_out;                   // (B,M,Cout)
  float* p2 = y + (size_t)BB * MM * COUT;      // (B,M,3) appended

  float* h       = (float*)d_ws;                            // (B*N, Cout) fp32
  int*   fpsIdx  = (int*)(h + (size_t)BB * NN * COUT);      // (B,M)
  int*   nbr     = fpsIdx + BB * MM;                        // (B,M,K)
  float* meanArr = (float*)(nbr + (size_t)BB * MM * KNN);   // (Cout,)
  float* rstdArr = meanArr + COUT;                          // (Cout,)

  fps_kernel<<<BB, 1024, 0, stream>>>(p1, fpsIdx, p2);
  gemm_wmma_kernel<<<(BB * NN) / 16, 256, 0, stream>>>(x, W, h);
  bnstats_kernel<<<COUT, 256, 0, stream>>>(h, meanArr, rstdArr);
  knn_kernel<<<BB * (MM / 128), 128, 0, stream>>>(p1, p2, nbr);
  gather_bn_max_kernel<<<BB * MM, 128, 0, stream>>>(h, nbr, meanArr, rstdArr,
                                                    gamma, beta, y);
}